// UmbrellaSurfaceConstructor_18253611008389
// MI455X (gfx1250) — compile-verified
//
#include <hip/hip_runtime.h>
#include <math.h>
#include <stdint.h>

typedef __attribute__((ext_vector_type(2))) float v2f;
typedef __attribute__((ext_vector_type(8))) float v8f;

typedef __attribute__((address_space(1))) int as1_int;   // global (AS1)
typedef __attribute__((address_space(3))) int as3_int;   // LDS (AS3)

#define B_   4
#define N_   8192
#define G_   8
#define C_   9
#define NCOL (B_ * G_ * N_)   /* 262144 columns (b,g,n) */
#define BN_EPS 1e-5f
#define PI_F 3.14159265358979323846f

#if __has_builtin(__builtin_amdgcn_global_load_async_to_lds_b32)
#define USE_ASYNC_LDS 1
#else
#define USE_ASYNC_LDS 0
#endif

// ---------------------------------------------------------------- zero
__global__ void zero_kernel(float* __restrict__ p, int count) {
    for (int i = blockIdx.x * blockDim.x + threadIdx.x; i < count;
         i += gridDim.x * blockDim.x)
        p[i] = 0.0f;
}

// ---------------------------------------------------------------- kNN
// One thread per (b, n). LDS-tiled candidate sweep via gfx1250 async
// memory->LDS loads (ASYNCcnt), top-9 insertion ladder in registers.
__global__ void knn_kernel(const float* __restrict__ center,
                           int* __restrict__ idxbuf) {
    const int blocksPerBatch = N_ / 256;
    int b = blockIdx.x / blocksPerBatch;
    int n = (blockIdx.x % blocksPerBatch) * 256 + threadIdx.x;
    const float* cb = center + (size_t)b * 3 * N_;
    float px = cb[n], py = cb[N_ + n], pz = cb[2 * N_ + n];

    float bd[9];
    int   bi[9];
#pragma unroll
    for (int k = 0; k < 9; ++k) { bd[k] = 3.4e38f; bi[k] = 0; }

    __shared__ float sx[256], sy[256], sz[256];
    for (int tile = 0; tile < N_; tile += 256) {
        __syncthreads();
#if USE_ASYNC_LDS
        {
            // global -> LDS async copy (tracked by ASYNCcnt), one b32/lane
            as1_int* g0 = (as1_int*)(uintptr_t)(cb + tile + threadIdx.x);
            as1_int* g1 = (as1_int*)(uintptr_t)(cb + N_ + tile + threadIdx.x);
            as1_int* g2 = (as1_int*)(uintptr_t)(cb + 2 * N_ + tile + threadIdx.x);
            // low 32 bits of a generic shared pointer == LDS byte offset
            as3_int* l0 = (as3_int*)(unsigned)(uintptr_t)&sx[threadIdx.x];
            as3_int* l1 = (as3_int*)(unsigned)(uintptr_t)&sy[threadIdx.x];
            as3_int* l2 = (as3_int*)(unsigned)(uintptr_t)&sz[threadIdx.x];
            __builtin_amdgcn_global_load_async_to_lds_b32(g0, l0, 0, 0);
            __builtin_amdgcn_global_load_async_to_lds_b32(g1, l1, 0, 0);
            __builtin_amdgcn_global_load_async_to_lds_b32(g2, l2, 0, 0);
            if (tile + 256 < N_)
                __builtin_prefetch(&cb[tile + 256 + threadIdx.x], 0, 0);
#if __has_builtin(__builtin_amdgcn_s_wait_asynccnt)
            __builtin_amdgcn_s_wait_asynccnt(0);
#else
            asm volatile("s_wait_asynccnt 0x0" ::: "memory");
#endif
        }
#else
        sx[threadIdx.x] = cb[tile + threadIdx.x];
        sy[threadIdx.x] = cb[N_ + tile + threadIdx.x];
        sz[threadIdx.x] = cb[2 * N_ + tile + threadIdx.x];
        if (tile + 256 < N_) {
            __builtin_prefetch(&cb[tile + 256 + threadIdx.x], 0, 0);
            __builtin_prefetch(&cb[N_ + tile + 256 + threadIdx.x], 0, 0);
            __builtin_prefetch(&cb[2 * N_ + tile + 256 + threadIdx.x], 0, 0);
        }
#endif
        __syncthreads();
        for (int j = 0; j < 256; ++j) {
            float dx = sx[j] - px, dy = sy[j] - py, dz = sz[j] - pz;
            float d = dx * dx + dy * dy + dz * dz;
            if (d < bd[8]) {
                int id = tile + j;
#pragma unroll
                for (int k = 8; k >= 1; --k) {
                    bool here = d < bd[k];
                    bool prev = d < bd[k - 1];
                    float nbv = prev ? bd[k - 1] : d;
                    int   niv = prev ? bi[k - 1] : id;
                    bd[k] = here ? nbv : bd[k];
                    bi[k] = here ? niv : bi[k];
                }
                if (d < bd[0]) { bd[0] = d; bi[0] = id; }
            }
        }
    }
    size_t base = ((size_t)b * N_ + n) * 8;
#pragma unroll
    for (int g = 0; g < 8; ++g) idxbuf[base + g] = bi[g + 1]; // drop self
}

// ---------------------------------------------------------------- features
__device__ inline void xyz2sphere(float x, float y, float z,
                                  float& rho, float& th, float& ph) {
    rho = sqrtf(x * x + y * y + z * z);
    float c = (rho == 0.0f) ? 1.0f : fminf(fmaxf(z / rho, -1.0f), 1.0f);
    th = (rho == 0.0f) ? 0.0f : acosf(c);
    ph = atan2f(y, x);
}

__global__ void feature_kernel(const float* __restrict__ center,
                               const int* __restrict__ idxbuf,
                               float* __restrict__ F) {
    int t = blockIdx.x * blockDim.x + threadIdx.x;
    if (t >= B_ * N_) return;
    int b = t / N_;
    int n = t % N_;
    const float* cb = center + (size_t)b * 3 * N_;
    float ox = cb[n], oy = cb[N_ + n], oz = cb[2 * N_ + n];

    float gx[8], gy[8], gz[8], ph[8];
    size_t ib = ((size_t)b * N_ + n) * 8;
#pragma unroll
    for (int g = 0; g < 8; ++g) {
        int id = idxbuf[ib + g];
        gx[g] = cb[id] - ox;
        gy[g] = cb[N_ + id] - oy;
        gz[g] = cb[2 * N_ + id] - oz;
        ph[g] = atan2f(gy[g], gx[g]);
    }
    // stable bubble sort by phi (strict >, so stable like jnp.argsort)
#pragma unroll
    for (int p = 0; p < 7; ++p) {
#pragma unroll
        for (int i = 0; i < 7; ++i) {
            if (ph[i] > ph[i + 1]) {
                float tf;
                tf = ph[i]; ph[i] = ph[i + 1]; ph[i + 1] = tf;
                tf = gx[i]; gx[i] = gx[i + 1]; gx[i + 1] = tf;
                tf = gy[i]; gy[i] = gy[i + 1]; gy[i + 1] = tf;
                tf = gz[i]; gz[i] = gz[i + 1]; gz[i + 1] = tf;
            }
        }
    }
    float ux[8], uy[8], uz[8], cx[8], cy[8], cz[8];
#pragma unroll
    for (int g = 0; g < 8; ++g) {
        int h = (g + 1) & 7;
        float ax = gx[g], ay = gy[g], az = gz[g];
        float bx = gx[h], by = gy[h], bz = gz[h];
        float nx = ay * bz - az * by;
        float ny = az * bx - ax * bz;
        float nz = ax * by - ay * bx;
        float len = sqrtf(nx * nx + ny * ny + nz * nz);
        ux[g] = nx / len; uy[g] = ny / len; uz[g] = nz / len; // 0/0 -> NaN
        cx[g] = (ax + bx) * (1.0f / 3.0f);
        cy[g] = (ay + by) * (1.0f / 3.0f);
        cz[g] = (az + bz) * (1.0f / 3.0f);
    }
    float s = (ux[0] > 0.0f) ? 1.0f : -1.0f;   // NaN -> -1, like reference
#pragma unroll
    for (int g = 0; g < 8; ++g) { ux[g] *= s; uy[g] *= s; uz[g] *= s; }

    // polar from PRE-replacement ctr (reference order)
    float pr[8], pt[8], pp[8];
#pragma unroll
    for (int g = 0; g < 8; ++g) {
        float rho, th, phv;
        xyz2sphere(cx[g], cy[g], cz[g], rho, th, phv);
        pr[g] = rho;
        pt[g] = th * (1.0f / PI_F);
        pp[g] = phv * (1.0f / (2.0f * PI_F)) + 0.5f;
    }
    bool nm[8];
#pragma unroll
    for (int g = 0; g < 8; ++g)
        nm[g] = (ux[g] != ux[g]) || (uy[g] != uy[g]) || (uz[g] != uz[g]);
    int first = 0; bool found = false;
#pragma unroll
    for (int g = 0; g < 8; ++g)
        if (!found && !nm[g]) { first = g; found = true; }
    float fx = ux[0], fy = uy[0], fz = uz[0];
    float fcx = cx[0], fcy = cy[0], fcz = cz[0];
#pragma unroll
    for (int g = 1; g < 8; ++g)
        if (first == g) { fx = ux[g]; fy = uy[g]; fz = uz[g];
                          fcx = cx[g]; fcy = cy[g]; fcz = cz[g]; }
#pragma unroll
    for (int g = 0; g < 8; ++g)
        if (nm[g]) { ux[g] = fx; uy[g] = fy; uz[g] = fz;
                     cx[g] = fcx; cy[g] = fcy; cz[g] = fcz; }

#pragma unroll
    for (int g = 0; g < 8; ++g) {
        size_t col = (size_t)(b * G_ + g) * N_ + n;
        F[0 * (size_t)NCOL + col] = cx[g];
        F[1 * (size_t)NCOL + col] = cy[g];
        F[2 * (size_t)NCOL + col] = cz[g];
        F[3 * (size_t)NCOL + col] = pr[g];
        F[4 * (size_t)NCOL + col] = pt[g];
        F[5 * (size_t)NCOL + col] = pp[g];
        F[6 * (size_t)NCOL + col] = ux[g];
        F[7 * (size_t)NCOL + col] = uy[g];
        F[8 * (size_t)NCOL + col] = uz[g];
    }
}

// ---------------------------------------------------------------- WMMA GEMM
// Y[0..8][col] = sum_k W[o][k] * X[k][col] (+bias). 9x9 padded to 16x12,
// three V_WMMA_F32_16X16X4_F32 per 16-column tile. X has 12 rows; rows 9..11
// are pre-zeroed K-padding. Bias is hoisted into registers before the loop.
template <bool HAS_BIAS>
__global__ void gemm_wmma_kernel(const float* __restrict__ W,
                                 const float* __restrict__ bias,
                                 const float* __restrict__ X,
                                 float* __restrict__ Y) {
    int lane   = threadIdx.x & 31;
    int wave   = (blockIdx.x * blockDim.x + threadIdx.x) >> 5;
    int nWaves = (gridDim.x * blockDim.x) >> 5;
    int m    = lane & 15;
    int koff = (lane & 16) ? 2 : 0;   // A: VGPR0/1 = K k0+koff, k0+koff+1

    v2f a[3];
#pragma unroll
    for (int t = 0; t < 3; ++t) {
        int k = t * 4 + koff;
        a[t].x = (m < C_ && k     < C_) ? W[m * C_ + k]     : 0.0f;
        a[t].y = (m < C_ && k + 1 < C_) ? W[m * C_ + k + 1] : 0.0f;
    }

    float badd[8];
#pragma unroll
    for (int r = 0; r < 8; ++r) badd[r] = 0.0f;
    if (HAS_BIAS) {
        if (lane < 16) {
#pragma unroll
            for (int r = 0; r < 8; ++r) badd[r] = bias[r];
        } else {
            badd[0] = bias[8];
        }
    }

    const int numTiles = NCOL / 16;
#pragma unroll 2
    for (int tile = wave; tile < numTiles; tile += nWaves) {
        int col = tile * 16 + (lane & 15);
        v8f c = {0.f, 0.f, 0.f, 0.f, 0.f, 0.f, 0.f, 0.f};
#pragma unroll
        for (int t = 0; t < 3; ++t) {
            int k = t * 4 + koff;
            v2f bf;
            bf.x = X[(size_t)k * NCOL + col];
            bf.y = X[(size_t)(k + 1) * NCOL + col];
            c = __builtin_amdgcn_wmma_f32_16x16x4_f32(
                    false, a[t], false, bf, (short)0, c, false, false);
        }
        // D layout: lanes 0-15 hold rows 0..7 in c[0..7]; lanes 16-31 rows 8..15
        if (lane < 16) {
#pragma unroll
            for (int r = 0; r < 8; ++r)
                Y[(size_t)r * NCOL + col] = c[r] + badd[r];
        } else {
            Y[(size_t)8 * NCOL + col] = c[0] + badd[0];
        }
    }
}

// ---------------------------------------------------------------- BN stats
__global__ void bn_stats_kernel(const float* __restrict__ X,
                                float* __restrict__ partials) {
    float ls[C_], lq[C_];
#pragma unroll
    for (int c = 0; c < C_; ++c) { ls[c] = 0.f; lq[c] = 0.f; }
    for (int col = blockIdx.x * blockDim.x + threadIdx.x; col < NCOL;
         col += gridDim.x * blockDim.x) {
#pragma unroll
        for (int c = 0; c < C_; ++c) {
            float v = X[(size_t)c * NCOL + col];
            ls[c] += v;
            lq[c] += v * v;
        }
    }
#pragma unroll
    for (int off = 16; off > 0; off >>= 1) {
#pragma unroll
        for (int c = 0; c < C_; ++c) {
            ls[c] += __shfl_down(ls[c], off, 32);
            lq[c] += __shfl_down(lq[c], off, 32);
        }
    }
    __shared__ float red[8][18];
    int wid = threadIdx.x >> 5, lane = threadIdx.x & 31;
    if (lane == 0) {
#pragma unroll
        for (int c = 0; c < C_; ++c) { red[wid][c] = ls[c]; red[wid][9 + c] = lq[c]; }
    }
    __syncthreads();
    if (threadIdx.x < 18) {
        float s = 0.f;
        for (int w = 0; w < (int)(blockDim.x >> 5); ++w) s += red[w][threadIdx.x];
        partials[blockIdx.x * 18 + threadIdx.x] = s;
    }
}

__global__ void bn_finalize_kernel(const float* __restrict__ partials,
                                   int nblocks, float* __restrict__ stats) {
    int j = threadIdx.x;
    if (j < C_) {
        float s = 0.f, q = 0.f;
        for (int b = 0; b < nblocks; ++b) {
            s += partials[b * 18 + j];
            q += partials[b * 18 + 9 + j];
        }
        float mean = s / (float)NCOL;
        float var  = q / (float)NCOL - mean * mean;
        stats[j]      = mean;
        stats[16 + j] = rsqrtf(var + BN_EPS);
    }
}

__global__ void bn_apply_kernel(const float* __restrict__ X,
                                const float* __restrict__ stats,
                                const float* __restrict__ gamma,
                                const float* __restrict__ beta,
                                float* __restrict__ Yout) {
    int total = C_ * NCOL;
    for (int i = blockIdx.x * blockDim.x + threadIdx.x; i < total;
         i += gridDim.x * blockDim.x) {
        int c = i / NCOL;
        float v = X[i];
        float r = (v - stats[c]) * stats[16 + c] * gamma[c] + beta[c];
        Yout[i] = fmaxf(r, 0.0f);
    }
}

// ---------------------------------------------------------------- group sum
__global__ void sum_group_kernel(const float* __restrict__ X,
                                 float* __restrict__ out) {
    int total = B_ * C_ * N_;
    for (int i = blockIdx.x * blockDim.x + threadIdx.x; i < total;
         i += gridDim.x * blockDim.x) {
        int n = i % N_;
        int o = (i / N_) % C_;
        int b = i / (N_ * C_);
        float s = 0.f;
#pragma unroll
        for (int g = 0; g < G_; ++g)
            s += X[(size_t)o * NCOL + (size_t)(b * G_ + g) * N_ + n];
        out[i] = s;
    }
}

// ---------------------------------------------------------------- launch
extern "C" void kernel_launch(void* const* d_in, const int* in_sizes, int n_in,
                              void* d_out, int out_size, void* d_ws, size_t ws_size,
                              hipStream_t stream) {
    const float* center = (const float*)d_in[0];
    const float* w1     = (const float*)d_in[1];
    const float* bn1_g  = (const float*)d_in[2];
    const float* bn1_b  = (const float*)d_in[3];
    const float* w2     = (const float*)d_in[4];
    const float* b2     = (const float*)d_in[5];
    const float* bn2_g  = (const float*)d_in[6];
    const float* bn2_b  = (const float*)d_in[7];
    const float* w3     = (const float*)d_in[8];
    const float* b3     = (const float*)d_in[9];
    (void)in_sizes; (void)n_in; (void)out_size; (void)ws_size;

    float* bufA     = (float*)d_ws;                       // [12][NCOL]
    float* bufB     = bufA + (size_t)12 * NCOL;           // [12][NCOL]
    float* stats    = bufB + (size_t)12 * NCOL;           // 64 floats (2 stages)
    float* partials = stats + 64;                         // 128 * 18
    int*   idxbuf   = (int*)(partials + 128 * 18);        // B*N*8 ints

    // zero the K-padding rows 9..11 of both ping-pong buffers (once per call)
    zero_kernel<<<64, 256, 0, stream>>>(bufA + (size_t)9 * NCOL, 3 * NCOL);
    zero_kernel<<<64, 256, 0, stream>>>(bufB + (size_t)9 * NCOL, 3 * NCOL);

    knn_kernel<<<(B_ * N_) / 256, 256, 0, stream>>>(center, idxbuf);
    feature_kernel<<<(B_ * N_) / 256, 256, 0, stream>>>(center, idxbuf, bufA);

    // stage 1: w1 * feat -> BN1 -> relu
    gemm_wmma_kernel<false><<<256, 256, 0, stream>>>(w1, nullptr, bufA, bufB);
    bn_stats_kernel<<<128, 256, 0, stream>>>(bufB, partials);
    bn_finalize_kernel<<<1, 32, 0, stream>>>(partials, 128, stats);
    bn_apply_kernel<<<512, 256, 0, stream>>>(bufB, stats, bn1_g, bn1_b, bufA);

    // stage 2: w2 * x + b2 -> BN2 -> relu
    gemm_wmma_kernel<true><<<256, 256, 0, stream>>>(w2, b2, bufA, bufB);
    bn_stats_kernel<<<128, 256, 0, stream>>>(bufB, partials);
    bn_finalize_kernel<<<1, 32, 0, stream>>>(partials, 128, stats + 32);
    bn_apply_kernel<<<512, 256, 0, stream>>>(bufB, stats + 32, bn2_g, bn2_b, bufA);

    // stage 3: w3 * x + b3 -> sum over group
    gemm_wmma_kernel<true><<<256, 256, 0, stream>>>(w3, b3, bufA, bufB);
    sum_group_kernel<<<288, 256, 0, stream>>>(bufB, (float*)d_out);
}